// OHEMCrossEntropy2D_40261023433177
// MI455X (gfx1250) — compile-verified
//
#include <hip/hip_runtime.h>
#include <cstdint>

// Problem constants (static per the reference)
#define BB   8
#define CC   19
#define HH   512
#define WW   1024
#define HW   (HH * WW)          // 524288
#define NPIX (BB * HW)          // 4194304
#define KSEL 2936012u           // max(int(0.7*NPIX), 100000)
#define TILE 1024               // pixels per block in the loss kernel
#define SLICES 4                // LDS histogram replication (contention relief)

// ---------------- CDNA5 async global->LDS path (guarded) ----------------
#if defined(__HIP_DEVICE_COMPILE__) && \
    __has_builtin(__builtin_amdgcn_global_load_async_to_lds_b128) && \
    __has_builtin(__builtin_amdgcn_s_wait_asynccnt)
#define HAVE_ASYNC 1
#else
#define HAVE_ASYNC 0
#endif

#if HAVE_ASYNC
// Builtin signature (from clang diagnostic): pointee is int __vector_size__(16),
// args are (AS1 ptr, AS3 ptr, imm offset, imm cpol).
typedef int v4i __attribute__((__vector_size__(4 * sizeof(int))));
typedef __attribute__((address_space(1))) v4i gv4i;
typedef __attribute__((address_space(3))) v4i lv4i;
__device__ __forceinline__ void async_cp16(const void* gp, void* lp) {
  // emits global_load_async_to_lds_b128 (tracked by ASYNCcnt)
  __builtin_amdgcn_global_load_async_to_lds_b128((gv4i*)gp, (lv4i*)lp, 0, 0);
}
__device__ __forceinline__ void async_wait0() {
  __builtin_amdgcn_s_wait_asynccnt(0);   // s_wait_asynccnt 0
}
#endif

// ---------------- Kernel 1: per-pixel CE loss + level-0 histogram ----------------
// Each block handles 1024 contiguous pixels of one batch image. The 19 channel
// rows (19 x 4KB) are staged into LDS via async copies (each thread copies the
// 16B it will later consume, so a per-wave s_wait_asynccnt suffices).
__global__ __launch_bounds__(256) void k_loss_hist0(
    const float* __restrict__ logits, const int* __restrict__ target,
    float* __restrict__ losses, unsigned* __restrict__ hist0) {
  __shared__ __align__(16) float sm[CC * TILE];     // 76 KB
  __shared__ unsigned h[SLICES * 2048];             // 32 KB
  const int t = threadIdx.x;
  for (int i = t; i < SLICES * 2048; i += 256) h[i] = 0;

  const size_t tile_start = (size_t)blockIdx.x * TILE;
  const int b   = (int)(tile_start / HW);
  const int hw0 = (int)(tile_start % HW);           // multiple of 1024
  const float* src = logits + (size_t)b * CC * HW + hw0 + t * 4;

#if HAVE_ASYNC
  #pragma unroll
  for (int c = 0; c < CC; ++c)
    async_cp16(src + (size_t)c * HW, &sm[c * TILE + t * 4]);
  async_wait0();
#else
  #pragma unroll
  for (int c = 0; c < CC; ++c)
    *(float4*)&sm[c * TILE + t * 4] = *(const float4*)(src + (size_t)c * HW);
#endif
  __syncthreads();   // histogram zeros visible; staged data is thread-local

  // pass 1: channel max (4 pixels per thread)
  float4 v = *(const float4*)&sm[t * 4];
  float m0 = v.x, m1 = v.y, m2 = v.z, m3 = v.w;
  #pragma unroll
  for (int c = 1; c < CC; ++c) {
    v = *(const float4*)&sm[c * TILE + t * 4];
    m0 = fmaxf(m0, v.x); m1 = fmaxf(m1, v.y);
    m2 = fmaxf(m2, v.z); m3 = fmaxf(m3, v.w);
  }
  // pass 2: sum of exp
  float s0 = 0.f, s1 = 0.f, s2 = 0.f, s3 = 0.f;
  #pragma unroll
  for (int c = 0; c < CC; ++c) {
    v = *(const float4*)&sm[c * TILE + t * 4];
    s0 += __expf(v.x - m0); s1 += __expf(v.y - m1);
    s2 += __expf(v.z - m2); s3 += __expf(v.w - m3);
  }
  const int4 tg = *(const int4*)(target + (size_t)b * HW + hw0 + t * 4);

  auto pix = [&](int tv, float m, float s, int i) -> float {
    unsigned ti = (unsigned)tv;
    const bool ig = (ti == 255u);                 // IGNORE_INDEX
    if (ti >= (unsigned)CC) ti = 0u;              // safe gather
    const float xt = sm[ti * TILE + t * 4 + i];
    const float l = m + __logf(s) - xt;           // logsumexp - x_target
    return ig ? 0.0f : fmaxf(l, 0.0f);            // loss >= 0 -> bits monotone
  };
  const float l0 = pix(tg.x, m0, s0, 0);
  const float l1 = pix(tg.y, m1, s1, 1);
  const float l2 = pix(tg.z, m2, s2, 2);
  const float l3 = pix(tg.w, m3, s3, 3);
  *(float4*)&losses[tile_start + t * 4] = make_float4(l0, l1, l2, l3);

  const unsigned slice = (unsigned)(t & (SLICES - 1)) * 2048u;
  atomicAdd(&h[slice + ((__float_as_uint(l0) >> 21) & 2047u)], 1u);
  atomicAdd(&h[slice + ((__float_as_uint(l1) >> 21) & 2047u)], 1u);
  atomicAdd(&h[slice + ((__float_as_uint(l2) >> 21) & 2047u)], 1u);
  atomicAdd(&h[slice + ((__float_as_uint(l3) >> 21) & 2047u)], 1u);
  __syncthreads();
  for (int i = t; i < 2048; i += 256) {
    const unsigned tot = h[i] + h[2048 + i] + h[4096 + i] + h[6144 + i];
    if (tot) atomicAdd(&hist0[i], tot);
  }
}

// ---------------- Radix-select: pick bin containing the k-th largest ----------------
// state[0..2]=b0,b1,b2  state[3]=k_rem  state[4]=cnt_gt  state[5]=threshold bits
__global__ __launch_bounds__(256) void k_select(
    const unsigned* __restrict__ hist, unsigned nbins, unsigned level,
    unsigned* __restrict__ state) {
  const int t = threadIdx.x;
  const unsigned bpt = nbins >> 8;                  // 8 or 4 bins/thread
  unsigned cnt[8];
  const unsigned base = (unsigned)t * bpt;
  unsigned local = 0;
  for (unsigned j = 0; j < bpt; ++j) { cnt[j] = hist[base + j]; local += cnt[j]; }
  __shared__ unsigned sc[256];
  sc[t] = local;
  __syncthreads();
  // Hillis-Steele suffix scan: sc[t] = sum over chunks t..255
  for (int d = 1; d < 256; d <<= 1) {
    const unsigned vv = (t + d < 256) ? sc[t + d] : 0u;
    __syncthreads();
    sc[t] += vv;
    __syncthreads();
  }
  const unsigned k    = (level == 0) ? KSEL : state[3];
  const unsigned suf  = sc[t];
  const unsigned sufn = (t < 255) ? sc[t + 1] : 0u;
  if (suf >= k && sufn < k) {                       // exactly one thread
    unsigned cum = sufn;                            // count strictly above my chunk
    for (int j = (int)bpt - 1; j >= 0; --j) {
      const unsigned c = cnt[j];
      if (cum + c >= k) {
        const unsigned bin = base + (unsigned)j;
        state[level] = bin;
        state[4] = state[4] + cum;                  // strictly-greater count
        state[3] = k - cum;                         // remaining to take in this bin
        if (level == 2)
          state[5] = (state[0] << 21) | (state[1] << 10) | bin;  // exact kth value
        break;
      }
      cum += c;
    }
  }
}

// ---------------- Refinement histograms over the selected bin ----------------
template <int LEVEL>
__global__ __launch_bounds__(256) void k_refine(
    const float* __restrict__ losses, const unsigned* __restrict__ state,
    unsigned* __restrict__ histo) {
  __shared__ unsigned h[2048];
  for (int i = threadIdx.x; i < 2048; i += 256) h[i] = 0;
  __syncthreads();
  const unsigned b0 = state[0], b1 = state[1];
  const unsigned pref = (LEVEL == 1) ? b0 : ((b0 << 11) | b1);
  const float4* L4 = (const float4*)losses;
  const unsigned n4 = NPIX / 4;
  for (unsigned idx = blockIdx.x * 256u + threadIdx.x; idx < n4;
       idx += gridDim.x * 256u) {
    const float4 v = L4[idx];
    const unsigned bt[4] = {__float_as_uint(v.x), __float_as_uint(v.y),
                            __float_as_uint(v.z), __float_as_uint(v.w)};
    #pragma unroll
    for (int j = 0; j < 4; ++j) {
      if (LEVEL == 1) {
        if ((bt[j] >> 21) == pref) atomicAdd(&h[(bt[j] >> 10) & 2047u], 1u);
      } else {
        if ((bt[j] >> 10) == pref) atomicAdd(&h[bt[j] & 1023u], 1u);
      }
    }
  }
  __syncthreads();
  const int nb = (LEVEL == 1) ? 2048 : 1024;
  for (int i = threadIdx.x; i < nb; i += 256)
    if (h[i]) atomicAdd(&histo[i], h[i]);
}

// ---------------- Sum of all elements strictly greater than threshold ----------------
__global__ __launch_bounds__(256) void k_sum(
    const float* __restrict__ losses, const unsigned* __restrict__ state,
    double* __restrict__ blockSums) {
  const unsigned tb = state[5];
  const float4* L4 = (const float4*)losses;
  float s = 0.f;
  const unsigned n4 = NPIX / 4;
  for (unsigned idx = blockIdx.x * 256u + threadIdx.x; idx < n4;
       idx += gridDim.x * 256u) {
    const float4 v = L4[idx];
    if (__float_as_uint(v.x) > tb) s += v.x;
    if (__float_as_uint(v.y) > tb) s += v.y;
    if (__float_as_uint(v.z) > tb) s += v.z;
    if (__float_as_uint(v.w) > tb) s += v.w;
  }
  #pragma unroll
  for (int o = 16; o > 0; o >>= 1) s += __shfl_xor(s, o, 32);   // wave32 reduce
  __shared__ float wsum[8];
  if ((threadIdx.x & 31) == 0) wsum[threadIdx.x >> 5] = s;
  __syncthreads();
  if (threadIdx.x == 0) {
    double d = 0.0;
    for (int i = 0; i < 8; ++i) d += (double)wsum[i];
    blockSums[blockIdx.x] = d;    // fixed slot -> deterministic final reduce
  }
}

// ---------------- Deterministic final reduction + mean ----------------
__global__ __launch_bounds__(256) void k_final(
    const unsigned* __restrict__ state, const double* __restrict__ blockSums,
    float* __restrict__ out) {
  __shared__ double sd[256];
  const int t = threadIdx.x;
  sd[t] = blockSums[t] + blockSums[t + 256] + blockSums[t + 512] + blockSums[t + 768];
  __syncthreads();
  for (int o = 128; o > 0; o >>= 1) {
    if (t < o) sd[t] += sd[t + o];
    __syncthreads();
  }
  if (t == 0) {
    const double sum_gt = sd[0];
    const unsigned cnt_gt = state[4];
    const float tv = __uint_as_float(state[5]);
    const double k_at = (double)(KSEL - cnt_gt);   // elements taken exactly at t
    out[0] = (float)((sum_gt + k_at * (double)tv) / (double)KSEL);
  }
}

// ---------------- Host launcher ----------------
extern "C" void kernel_launch(void* const* d_in, const int* in_sizes, int n_in,
                              void* d_out, int out_size, void* d_ws, size_t ws_size,
                              hipStream_t stream) {
  (void)in_sizes; (void)n_in; (void)out_size; (void)ws_size;
  const float* logits = (const float*)d_in[0];
  const int*   target = (const int*)d_in[1];

  // Workspace layout (needs ~16.81 MB):
  // [0, 16.78MB)   loss array (float[NPIX])
  // then hist0[2048], hist1[2048], hist2[1024], state[8] (u32), blockSums[1024] (f64)
  char* ws = (char*)d_ws;
  float*    losses = (float*)ws;
  unsigned* hist0  = (unsigned*)(ws + (size_t)NPIX * 4);
  unsigned* hist1  = hist0 + 2048;
  unsigned* hist2  = hist1 + 2048;
  unsigned* state  = hist2 + 1024;
  double*   bsums  = (double*)(ws + (size_t)NPIX * 4 + (2048 + 2048 + 1024 + 8) * 4);

  (void)hipMemsetAsync(hist0, 0, (2048 + 2048 + 1024 + 8) * sizeof(unsigned), stream);
  k_loss_hist0<<<NPIX / TILE, 256, 0, stream>>>(logits, target, losses, hist0);
  k_select    <<<1,    256, 0, stream>>>(hist0, 2048, 0, state);
  k_refine<1> <<<1024, 256, 0, stream>>>(losses, state, hist1);
  k_select    <<<1,    256, 0, stream>>>(hist1, 2048, 1, state);
  k_refine<2> <<<1024, 256, 0, stream>>>(losses, state, hist2);
  k_select    <<<1,    256, 0, stream>>>(hist2, 1024, 2, state);
  k_sum       <<<1024, 256, 0, stream>>>(losses, state, bsums);
  k_final     <<<1,    256, 0, stream>>>(state, bsums, (float*)d_out);
}